// GATEncoder_9620726743402
// MI455X (gfx1250) — compile-verified
//
#include <hip/hip_runtime.h>
#include <hip/hip_bf16.h>

#define N_NODES   50000
#define N_EDGES   800000
#define IN_CH     128
#define OUT_CH    64
#define HEADS     2
#define NEG_SLOPE 0.2f
#define E_TOTAL   (N_EDGES + N_NODES)
#define GEMM_WAVES 4          // waves per block; share one LDS B-panel, cover 4 M-tiles
#define MAX_K      256        // largest K used (layers 2/3)

typedef float v2f __attribute__((ext_vector_type(2)));
typedef float v8f __attribute__((ext_vector_type(8)));

#if defined(__gfx1250__) && __has_builtin(__builtin_amdgcn_wmma_f32_16x16x4_f32)
#define HAVE_WMMA_F32X4 1
#endif

// ---------------------------------------------------------------------------
// GEMM: C[M,N] = A[M,K] * B[K,N], row-major. M%16==0, N%64==0, K%4==0, K<=256.
// Block = 4 waves. The block stages its 64-column B panel in LDS, swizzled
// into WMMA fragment order: lds[p*64+c] = (B[2p, n0+c], B[2p+1, n0+c]).
// Inner loop per K-step of 4: 1x global b64 (A frag) + 4x ds b64 (B frags)
// feeding 4x V_WMMA_F32_16X16X4_F32 (16x64 strip per wave).
// Fragment layouts (CDNA5 ISA 7.12.2):
//   A 16x4 : lane holds row (lane&15), VGPR v holds K = v + 2*(lane>>4)
//   B 4x16 : lane holds col (lane&15), VGPR v holds K = v + 2*(lane>>4)
//   C 16x16: VGPR v -> row = v + 8*(lane>>4), col = lane&15
// ---------------------------------------------------------------------------
__global__ void __launch_bounds__(32 * GEMM_WAVES)
gemm_wmma_f32(const float* __restrict__ A,
              const float* __restrict__ B,
              float* __restrict__ C,
              int M, int N, int K) {
    __shared__ float2 bpanel[(MAX_K / 2) * 64];          // 64 KB of 320 KB/WGP

    const int lane = threadIdx.x & 31;
    const int wv   = threadIdx.x >> 5;
    const int tilesN = N >> 6;
    const int bm = blockIdx.x / tilesN;
    const int n0 = (blockIdx.x % tilesN) << 6;

    // Cooperative, coalesced fill of the swizzled B panel.
    {
        float* lp = (float*)bpanel;
        const int total = K * 64;
        for (int i = threadIdx.x; i < total; i += 32 * GEMM_WAVES) {
            const int k = i >> 6;
            const int c = i & 63;
            lp[((((k >> 1) << 6) + c) << 1) + (k & 1)] = B[(size_t)k * N + n0 + c];
        }
    }
    __syncthreads();

    // Clamp trailing waves onto the last M-tile: redundant identical compute
    // and identical stores keep control flow convergent (EXEC all-ones).
    const int maxTileM = (M >> 4) - 1;
    int tm = bm * GEMM_WAVES + wv;
    if (tm > maxTileM) tm = maxTileM;
    const int m0 = tm << 4;
    const int r  = lane & 15;
    const int hi = lane >> 4;
    const int rowBase = m0 + (hi << 3);

#ifdef HAVE_WMMA_F32X4
    const int kk = hi << 1;
    v8f acc0 = {}, acc1 = {}, acc2 = {}, acc3 = {};
    const float* arow = A + (size_t)(m0 + r) * K;
#pragma unroll 4
    for (int k0 = 0; k0 < K; k0 += 4) {
        v2f a;
        a.x = arow[k0 + kk];
        a.y = arow[k0 + kk + 1];
        const float2* bp = bpanel + (((k0 >> 1) + hi) << 6) + r;
        const float2 f0 = bp[0];
        const float2 f1 = bp[16];
        const float2 f2 = bp[32];
        const float2 f3 = bp[48];
        v2f b0; b0.x = f0.x; b0.y = f0.y;
        v2f b1; b1.x = f1.x; b1.y = f1.y;
        v2f b2; b2.x = f2.x; b2.y = f2.y;
        v2f b3; b3.x = f3.x; b3.y = f3.y;
        acc0 = __builtin_amdgcn_wmma_f32_16x16x4_f32(false, a, false, b0, (short)0, acc0, false, false);
        acc1 = __builtin_amdgcn_wmma_f32_16x16x4_f32(false, a, false, b1, (short)0, acc1, false, false);
        acc2 = __builtin_amdgcn_wmma_f32_16x16x4_f32(false, a, false, b2, (short)0, acc2, false, false);
        acc3 = __builtin_amdgcn_wmma_f32_16x16x4_f32(false, a, false, b3, (short)0, acc3, false, false);
    }
#pragma unroll
    for (int v = 0; v < 8; ++v) {
        float* cp = C + (size_t)(rowBase + v) * N + n0 + r;
        cp[0]  = acc0[v];
        cp[16] = acc1[v];
        cp[32] = acc2[v];
        cp[48] = acc3[v];
    }
#else
    // Scalar fallback (host pass / builtin unavailable): same tile assignment.
    for (int v = 0; v < 8; ++v) {
        const float* arow2 = A + (size_t)(rowBase + v) * K;
        for (int t = 0; t < 4; ++t) {
            const int cidx = n0 + (t << 4) + r;
            float s = 0.f;
            for (int k = 0; k < K; ++k) s += arow2[k] * B[(size_t)k * N + cidx];
            C[(size_t)(rowBase + v) * N + cidx] = s;
        }
    }
#endif
}

// ---------------------------------------------------------------------------
// alpha_src[n,h] = sum_c h[n,h,c]*a_src[h,c];  alpha_dst likewise.
// One wave32 per (node, head); lane-strided loads + shuffle reduction.
// ---------------------------------------------------------------------------
__global__ void node_alpha_kernel(const float* __restrict__ h,
                                  const float* __restrict__ a_src,
                                  const float* __restrict__ a_dst,
                                  float* __restrict__ asrc,
                                  float* __restrict__ adst,
                                  int C) {
    const int wave = blockIdx.x * (blockDim.x >> 5) + (threadIdx.x >> 5);
    const int lane = threadIdx.x & 31;
    if (wave >= N_NODES * HEADS) return;
    const int node = wave / HEADS;
    const int hd   = wave % HEADS;
    const float* row = h + ((size_t)node * HEADS + hd) * C;
    const float* as  = a_src + (size_t)hd * C;
    const float* ad  = a_dst + (size_t)hd * C;
    float s0 = 0.f, s1 = 0.f;
    for (int c = lane; c < C; c += 32) {
        const float v = row[c];
        s0 += v * as[c];
        s1 += v * ad[c];
    }
    for (int off = 16; off > 0; off >>= 1) {
        s0 += __shfl_xor(s0, off, 32);
        s1 += __shfl_xor(s1, off, 32);
    }
    if (lane == 0) { asrc[wave] = s0; adst[wave] = s1; }
}

// Monotonic float <-> uint key so unsigned atomicMax == float max.
__device__ __forceinline__ unsigned fkey(float v) {
    unsigned u = __float_as_uint(v);
    return (u & 0x80000000u) ? ~u : (u | 0x80000000u);
}
__device__ __forceinline__ float funkey(unsigned k) {
    unsigned u = (k & 0x80000000u) ? (k ^ 0x80000000u) : ~k;
    return __uint_as_float(u);
}

__device__ __forceinline__ void edge_nodes(const int* __restrict__ ei, int e, int& s, int& d) {
    if (e < N_EDGES) { s = ei[e]; d = ei[N_EDGES + e]; }
    else             { s = e - N_EDGES; d = s; }          // appended self-loops
}

__global__ void init_softmax_kernel(unsigned* __restrict__ mk, float* __restrict__ dn) {
    const int i = blockIdx.x * blockDim.x + threadIdx.x;
    if (i >= N_NODES * HEADS) return;
    mk[i] = 0x007FFFFFu;   // fkey(-inf)
    dn[i] = 0.f;
}

__global__ void edge_max_kernel(const int* __restrict__ ei,
                                const float* __restrict__ asrc,
                                const float* __restrict__ adst,
                                unsigned* __restrict__ mk) {
    const int e = blockIdx.x * blockDim.x + threadIdx.x;
    if (e >= E_TOTAL) return;
    int s, d; edge_nodes(ei, e, s, d);
#pragma unroll
    for (int h = 0; h < HEADS; ++h) {
        float v = asrc[s * HEADS + h] + adst[d * HEADS + h];
        v = v > 0.f ? v : NEG_SLOPE * v;
        atomicMax(&mk[d * HEADS + h], fkey(v));
    }
}

__global__ void edge_expsum_kernel(const int* __restrict__ ei,
                                   const float* __restrict__ asrc,
                                   const float* __restrict__ adst,
                                   const unsigned* __restrict__ mk,
                                   float* __restrict__ dn,
                                   float* __restrict__ ew) {
    const int e = blockIdx.x * blockDim.x + threadIdx.x;
    if (e >= E_TOTAL) return;
    int s, d; edge_nodes(ei, e, s, d);
#pragma unroll
    for (int h = 0; h < HEADS; ++h) {
        float v = asrc[s * HEADS + h] + adst[d * HEADS + h];
        v = v > 0.f ? v : NEG_SLOPE * v;
        const float m = funkey(mk[d * HEADS + h]);
        const float w = __expf(v - m);
        ew[(size_t)e * HEADS + h] = w;
        atomicAdd(&dn[d * HEADS + h], w);
    }
}

__global__ void fill_f32_kernel(float* __restrict__ p, float v, long long n) {
    const long long i = (long long)blockIdx.x * blockDim.x + threadIdx.x;
    if (i < n) p[i] = v;
}

// One thread per (edge, head, 4-channel group): gather h[src], scatter-add to acc[dst].
__global__ void edge_aggregate_kernel(const int* __restrict__ ei,
                                      const float* __restrict__ ew,
                                      const float* __restrict__ dn,
                                      const float* __restrict__ h,
                                      float* __restrict__ acc,
                                      int C) {
    const int CG = C >> 2;
    const long long total = (long long)E_TOTAL * HEADS * CG;
    const long long t = (long long)blockIdx.x * blockDim.x + threadIdx.x;
    if (t >= total) return;
    const int cg = (int)(t % CG);
    const long long t2 = t / CG;
    const int hd = (int)(t2 % HEADS);
    const int e  = (int)(t2 / HEADS);
    int s, d; edge_nodes(ei, e, s, d);
    const float alpha = ew[(size_t)e * HEADS + hd] / (dn[d * HEADS + hd] + 1e-16f);
    const float4 hv = *(const float4*)(h + ((size_t)s * HEADS + hd) * C + (cg << 2));
    float* op = acc + ((size_t)d * HEADS + hd) * C + (cg << 2);
    atomicAdd(op + 0, alpha * hv.x);
    atomicAdd(op + 1, alpha * hv.y);
    atomicAdd(op + 2, alpha * hv.z);
    atomicAdd(op + 3, alpha * hv.w);
}

__global__ void finalize_kernel(const float* __restrict__ src,
                                float* __restrict__ dst,
                                const float* __restrict__ bias,
                                long long total, int HC, int do_elu) {
    const long long i = (long long)blockIdx.x * blockDim.x + threadIdx.x;
    if (i >= total) return;
    float v = src[i] + bias[(int)(i % HC)];
    if (do_elu) v = v > 0.f ? v : __expf(v) - 1.0f;
    dst[i] = v;
}

// ---------------------------------------------------------------------------
static void run_gat_layer(const float* X, const float* W,
                          const float* a_src, const float* a_dst, const float* bias,
                          const int* ei,
                          float* hbuf, float* accbuf, float* outbuf,
                          float* asrc, float* adst, unsigned* mk, float* dn, float* ew,
                          int K, int C, int do_elu, hipStream_t stream) {
    const int N = HEADS * C;
    const int tilesN = N / 64;
    const int mBlocks = ((N_NODES / 16) + GEMM_WAVES - 1) / GEMM_WAVES;
    gemm_wmma_f32<<<mBlocks * tilesN, 32 * GEMM_WAVES, 0, stream>>>(X, W, hbuf, N_NODES, N, K);

    const int waves = N_NODES * HEADS;
    node_alpha_kernel<<<(waves + 7) / 8, 256, 0, stream>>>(hbuf, a_src, a_dst, asrc, adst, C);
    init_softmax_kernel<<<(N_NODES * HEADS + 255) / 256, 256, 0, stream>>>(mk, dn);
    edge_max_kernel<<<(E_TOTAL + 255) / 256, 256, 0, stream>>>(ei, asrc, adst, mk);
    edge_expsum_kernel<<<(E_TOTAL + 255) / 256, 256, 0, stream>>>(ei, asrc, adst, mk, dn, ew);

    const long long accn = (long long)N_NODES * N;
    fill_f32_kernel<<<(int)((accn + 255) / 256), 256, 0, stream>>>(accbuf, 0.f, accn);
    const long long aggt = (long long)E_TOTAL * HEADS * (C >> 2);
    edge_aggregate_kernel<<<(int)((aggt + 255) / 256), 256, 0, stream>>>(ei, ew, dn, hbuf, accbuf, C);
    finalize_kernel<<<(int)((accn + 255) / 256), 256, 0, stream>>>(accbuf, outbuf, bias, accn, N, do_elu);
}

extern "C" void kernel_launch(void* const* d_in, const int* in_sizes, int n_in,
                              void* d_out, int out_size, void* d_ws, size_t ws_size,
                              hipStream_t stream) {
    (void)in_sizes; (void)n_in; (void)out_size; (void)ws_size;
    const float* x     = (const float*)d_in[0];
    const int*   ei    = (const int*)d_in[1];
    const float* W1    = (const float*)d_in[2];
    const float* a_s1  = (const float*)d_in[3];
    const float* a_d1  = (const float*)d_in[4];
    const float* b1    = (const float*)d_in[5];
    const float* W_mu  = (const float*)d_in[6];
    const float* a_smu = (const float*)d_in[7];
    const float* a_dmu = (const float*)d_in[8];
    const float* b_mu  = (const float*)d_in[9];
    const float* W_ls  = (const float*)d_in[10];
    const float* a_sls = (const float*)d_in[11];
    const float* a_dls = (const float*)d_in[12];
    const float* b_ls  = (const float*)d_in[13];

    float* ws = (float*)d_ws;
    size_t off = 0;
    float* hbuf = ws + off; off += (size_t)N_NODES * 256;   // h (max width 256), reused per layer
    float* x1   = ws + off; off += (size_t)N_NODES * 256;   // ELU(layer1) output
    float* acc1 = ws + off; off += (size_t)N_NODES * 256;   // layer1 accumulator
    float* asrc = ws + off; off += (size_t)N_NODES * HEADS;
    float* adst = ws + off; off += (size_t)N_NODES * HEADS;
    float* dn   = ws + off; off += (size_t)N_NODES * HEADS;
    unsigned* mk = (unsigned*)(ws + off); off += (size_t)N_NODES * HEADS;
    float* ew   = ws + off; off += (size_t)E_TOTAL * HEADS;

    float* mu = (float*)d_out;
    float* ls = mu + (size_t)N_NODES * HEADS * OUT_CH;

    // Layer 1: 128 -> (H=2, C=128), ELU
    run_gat_layer(x, W1, a_s1, a_d1, b1, ei, hbuf, acc1, x1,
                  asrc, adst, mk, dn, ew, IN_CH, 2 * OUT_CH, 1, stream);
    // mu head: 256 -> (H=2, C=64), accumulate straight into d_out
    run_gat_layer(x1, W_mu, a_smu, a_dmu, b_mu, ei, hbuf, mu, mu,
                  asrc, adst, mk, dn, ew, 2 * OUT_CH * HEADS, OUT_CH, 0, stream);
    // logstd head
    run_gat_layer(x1, W_ls, a_sls, a_dls, b_ls, ei, hbuf, ls, ls,
                  asrc, adst, mk, dn, ew, 2 * OUT_CH * HEADS, OUT_CH, 0, stream);
}